// EMDLoss_61649960567417
// MI455X (gfx1250) — compile-verified
//
#include <hip/hip_runtime.h>

typedef __attribute__((ext_vector_type(2))) float v2f;
typedef __attribute__((ext_vector_type(8))) float v8f;

// Problem constants (reference: template/source [32,1024,3], eps=0.02, 50 iters)
constexpr int   BB        = 32;
constexpr int   NN        = 1024;
constexpr float EPSR      = 0.02f;
constexpr float INV_EPS   = 50.0f;
constexpr float LOG_MARG  = -6.93147180559945f;   // log(1/N)
constexpr int   ITERS     = 50;

constexpr int BLOCK          = 256;               // 8 wave32
constexpr int WAVES          = BLOCK / 32;
constexpr int ROWS_PER_WAVE  = 16;
constexpr int ROWS_PER_BLOCK = WAVES * ROWS_PER_WAVE;   // 128
constexpr int ROWBLOCKS      = NN / ROWS_PER_BLOCK;     // 8

__global__ void emd_init_duals(float* __restrict__ fg) {
    int i = blockIdx.x * blockDim.x + threadIdx.x;
    if (i < 2 * BB * NN) fg[i] = 0.0f;
}

// One Sinkhorn half-update:
//   dual_out[M] = eps*(log_marg - logsumexp_j( (dual_in[j] - C(rowPts[M], colPts[j]))/eps ))
// Symmetric: f-update = (rows=template, cols=source, in=g); g-update = swapped.
__global__ __launch_bounds__(BLOCK) void sinkhorn_half(
    const float* __restrict__ rowPts,   // [N,3] for batch b (rows / output side)
    const float* __restrict__ colPts,   // [N,3] for batch b (cols / reduced side)
    const float* __restrict__ dual_in,  // [B,N] over cols
    float*       __restrict__ dual_out) // [B,N] over rows
{
    __shared__ float sx[NN], sy[NN], sz[NN], sn2[NN], sdual[NN];

    const int b    = blockIdx.y;
    const int tid  = threadIdx.x;
    const int lane = tid & 31;
    const int wave = tid >> 5;

    const float* pB  = colPts  + (size_t)b * NN * 3;
    const float* din = dual_in + (size_t)b * NN;
    for (int j = tid; j < NN; j += BLOCK) {
        float x = pB[3*j], y = pB[3*j+1], z = pB[3*j+2];
        sx[j] = x; sy[j] = y; sz[j] = z;
        sn2[j] = x*x + y*y + z*z;
        sdual[j] = din[j];
    }
    __syncthreads();

    const int    row0 = blockIdx.x * ROWS_PER_BLOCK + wave * ROWS_PER_WAVE;
    const float* pA   = rowPts + (size_t)b * NN * 3;

    // A fragment (16x4 f32): lanes0-15 hold M=lane (K=0,1)=(x,y); lanes16-31 hold
    // M=lane-16 (K=2,3)=(z,0). Scale by -2 so WMMA yields -2*t.s directly.
    const int aM = row0 + (lane & 15);
    v2f afrag;
    if (lane < 16) { afrag.x = -2.0f * pA[3*aM];     afrag.y = -2.0f * pA[3*aM+1]; }
    else           { afrag.x = -2.0f * pA[3*aM + 2]; afrag.y = 0.0f; }

    // Row norms for this lane's 8 D slots: M = mbase + r, mbase = row0 + 8*(lane>=16)
    const int mbase = row0 + ((lane >> 4) << 3);
    float tn2[8];
    #pragma unroll
    for (int r = 0; r < 8; ++r) {
        float x = pA[3*(mbase+r)], y = pA[3*(mbase+r)+1], z = pA[3*(mbase+r)+2];
        tn2[r] = x*x + y*y + z*z;
    }

    float mx[8], sm[8];
    #pragma unroll
    for (int r = 0; r < 8; ++r) { mx[r] = -3.0e38f; sm[r] = 0.0f; }

    const int col = lane & 15;
    for (int j0 = 0; j0 < NN; j0 += 16) {
        const int jc = j0 + col;
        // B fragment (4x16 f32): lanes0-15 rows K=0,1=(x,y); lanes16-31 rows K=2,3=(z,0)
        v2f bfrag;
        if (lane < 16) { bfrag.x = sx[jc]; bfrag.y = sy[jc]; }
        else           { bfrag.x = sz[jc]; bfrag.y = 0.0f;   }

        v8f acc = {0.f,0.f,0.f,0.f,0.f,0.f,0.f,0.f};
        acc = __builtin_amdgcn_wmma_f32_16x16x4_f32(false, afrag, false, bfrag,
                                                    (short)0, acc, false, false);
        const float cn2 = sn2[jc];
        const float gj  = sdual[jc];
        #pragma unroll
        for (int r = 0; r < 8; ++r) {
            float d2 = tn2[r] + cn2 + acc[r];          // |t|^2+|s|^2-2t.s
            float c  = __builtin_amdgcn_sqrtf(d2 + 1e-12f);
            float z  = (gj - c) * INV_EPS;
            float mn = fmaxf(mx[r], z);                // branchless streaming LSE
            sm[r] = sm[r] * __expf(mx[r] - mn) + __expf(z - mn);
            mx[r] = mn;
        }
    }

    // Row-wise LSE completion: combine the 16 lanes of each half-wave.
    #pragma unroll
    for (int r = 0; r < 8; ++r) {
        float m = mx[r], s = sm[r];
        #pragma unroll
        for (int msk = 8; msk >= 1; msk >>= 1) {
            float mo = __shfl_xor(m, msk, 32);
            float so = __shfl_xor(s, msk, 32);
            float mn = fmaxf(m, mo);
            s = s * __expf(m - mn) + so * __expf(mo - mn);
            m = mn;
        }
        if ((lane & 15) == 0) {
            float lse = m + __logf(s);
            dual_out[(size_t)b * NN + (mbase + r)] = EPSR * (LOG_MARG - lse);
        }
    }
}

// Partial sums of <P, C> = sum exp((f_i+g_j-C_ij)/eps) * C_ij
__global__ __launch_bounds__(BLOCK) void emd_loss_partial(
    const float* __restrict__ tpts, const float* __restrict__ spts,
    const float* __restrict__ fdual, const float* __restrict__ gdual,
    float* __restrict__ partials)
{
    __shared__ float sx[NN], sy[NN], sz[NN], sn2[NN], sdual[NN];
    __shared__ float wsum[WAVES];

    const int b    = blockIdx.y;
    const int tid  = threadIdx.x;
    const int lane = tid & 31;
    const int wave = tid >> 5;

    const float* pB  = spts  + (size_t)b * NN * 3;
    const float* din = gdual + (size_t)b * NN;
    for (int j = tid; j < NN; j += BLOCK) {
        float x = pB[3*j], y = pB[3*j+1], z = pB[3*j+2];
        sx[j] = x; sy[j] = y; sz[j] = z;
        sn2[j] = x*x + y*y + z*z;
        sdual[j] = din[j];
    }
    __syncthreads();

    const int    row0 = blockIdx.x * ROWS_PER_BLOCK + wave * ROWS_PER_WAVE;
    const float* pA   = tpts + (size_t)b * NN * 3;

    const int aM = row0 + (lane & 15);
    v2f afrag;
    if (lane < 16) { afrag.x = -2.0f * pA[3*aM];     afrag.y = -2.0f * pA[3*aM+1]; }
    else           { afrag.x = -2.0f * pA[3*aM + 2]; afrag.y = 0.0f; }

    const int mbase = row0 + ((lane >> 4) << 3);
    float tn2[8], fi[8];
    #pragma unroll
    for (int r = 0; r < 8; ++r) {
        float x = pA[3*(mbase+r)], y = pA[3*(mbase+r)+1], z = pA[3*(mbase+r)+2];
        tn2[r] = x*x + y*y + z*z;
        fi[r]  = fdual[(size_t)b * NN + (mbase + r)];
    }

    float total = 0.0f;
    const int col = lane & 15;
    for (int j0 = 0; j0 < NN; j0 += 16) {
        const int jc = j0 + col;
        v2f bfrag;
        if (lane < 16) { bfrag.x = sx[jc]; bfrag.y = sy[jc]; }
        else           { bfrag.x = sz[jc]; bfrag.y = 0.0f;   }

        v8f acc = {0.f,0.f,0.f,0.f,0.f,0.f,0.f,0.f};
        acc = __builtin_amdgcn_wmma_f32_16x16x4_f32(false, afrag, false, bfrag,
                                                    (short)0, acc, false, false);
        const float cn2 = sn2[jc];
        const float gj  = sdual[jc];
        #pragma unroll
        for (int r = 0; r < 8; ++r) {
            float d2 = tn2[r] + cn2 + acc[r];
            float c  = __builtin_amdgcn_sqrtf(d2 + 1e-12f);
            float p  = __expf((fi[r] + gj - c) * INV_EPS);
            total += p * c;
        }
    }

    #pragma unroll
    for (int msk = 16; msk >= 1; msk >>= 1) total += __shfl_xor(total, msk, 32);
    if (lane == 0) wsum[wave] = total;
    __syncthreads();
    if (tid == 0) {
        float t = 0.0f;
        #pragma unroll
        for (int w = 0; w < WAVES; ++w) t += wsum[w];
        partials[blockIdx.y * gridDim.x + blockIdx.x] = t;
    }
}

__global__ void emd_reduce(const float* __restrict__ partials,
                           float* __restrict__ out, int n, float scale) {
    __shared__ float sh[256];
    int tid = threadIdx.x;
    sh[tid] = (tid < n) ? partials[tid] : 0.0f;
    __syncthreads();
    for (int s = 128; s > 0; s >>= 1) {
        if (tid < s) sh[tid] += sh[tid + s];
        __syncthreads();
    }
    if (tid == 0) out[0] = sh[0] * scale;
}

extern "C" void kernel_launch(void* const* d_in, const int* in_sizes, int n_in,
                              void* d_out, int out_size, void* d_ws, size_t ws_size,
                              hipStream_t stream) {
    const float* tmpl = (const float*)d_in[0];   // [32,1024,3]
    const float* srcp = (const float*)d_in[1];   // [32,1024,3]
    float* out = (float*)d_out;

    float* f        = (float*)d_ws;              // [B,N]
    float* g        = f + (size_t)BB * NN;       // [B,N]
    float* partials = g + (size_t)BB * NN;       // [ROWBLOCKS*B] = 256

    emd_init_duals<<<(2 * BB * NN + 255) / 256, 256, 0, stream>>>(f);

    dim3 grid(ROWBLOCKS, BB);   // (8, 32)
    for (int it = 0; it < ITERS; ++it) {
        // f_i = eps*(logm - lse_j (g_j - C_ij)/eps)
        sinkhorn_half<<<grid, BLOCK, 0, stream>>>(tmpl, srcp, g, f);
        // g_j = eps*(logm - lse_i (f_i - C_ij)/eps)   (uses fresh f)
        sinkhorn_half<<<grid, BLOCK, 0, stream>>>(srcp, tmpl, f, g);
    }

    emd_loss_partial<<<grid, BLOCK, 0, stream>>>(tmpl, srcp, f, g, partials);
    emd_reduce<<<1, 256, 0, stream>>>(partials, out, ROWBLOCKS * BB,
                                      1.0f / (float)(BB * NN));
}